// PNANet_58205396795402
// MI455X (gfx1250) — compile-verified
//
#include <hip/hip_runtime.h>
#include <hip/hip_bf16.h>

#define N_NODES 50000
#define N_EDGES 800000
#define AVG_DEG_LOG 2.8332133440562162f
#define BN_EPS 1e-5f

typedef float v2f __attribute__((ext_vector_type(2)));
typedef float v8f __attribute__((ext_vector_type(8)));

// CDNA5 f32 WMMA: D(16x16,f32) = A(16x4,f32) x B(4x16,f32) + C
__device__ __forceinline__ v8f wmma_f32_k4(v2f a, v2f b, v8f c) {
  return __builtin_amdgcn_wmma_f32_16x16x4_f32(false, a, false, b, (short)0, c,
                                               false, false);
}

// ---------------------------------------------------------------------------
// Generic C[M x 64] = In[M x 64] @ W[64 x 64] (+ bias). One block = 16 rows,
// 4 waves each own a 16-col tile. K looped in steps of 4 via v_wmma_f32_16x16x4.
// ---------------------------------------------------------------------------
__global__ void __launch_bounds__(128) gemm64(
    const float* __restrict__ In, int ldi, const float* __restrict__ W,
    const float* __restrict__ bias, float* __restrict__ Out, int ldo) {
  __shared__ float Hs[16 * 68];   // padded: avoid bank conflicts on column reads
  __shared__ float Ws[64 * 64];
  const int tid = threadIdx.x;
  const int wave = tid >> 5, lane = tid & 31;
  const int m0 = blockIdx.x * 16;

  for (int i = tid; i < 16 * 64; i += 128) {
    int r = i >> 6, c = i & 63;
    Hs[r * 68 + c] = In[(m0 + r) * ldi + c];
  }
  for (int i = tid; i < 64 * 64; i += 128) Ws[i] = W[i];
  __syncthreads();

  const int n = lane & 15;        // M for A-frag, N for B/D-frag
  const int kh = lane >> 4;       // K-half selector
  const int col = wave * 16 + n;
  v8f acc = {};
#pragma unroll 4
  for (int k0 = 0; k0 < 64; k0 += 4) {
    const int ka = k0 + 2 * kh;
    v2f a, b;
    a.x = Hs[n * 68 + ka];
    a.y = Hs[n * 68 + ka + 1];
    b.x = Ws[ka * 64 + col];
    b.y = Ws[(ka + 1) * 64 + col];
    acc = wmma_f32_k4(a, b, acc);
  }
  const float bc = bias ? bias[col] : 0.0f;
#pragma unroll
  for (int r = 0; r < 8; ++r) {
    const int row = r + 8 * kh;
    Out[(m0 + row) * ldo + col] = acc[r] + bc;
  }
}

// ---------------------------------------------------------------------------
// Fused post GEMM: tmp = x@W0 + agg@W1 + s1*(agg@W2) + s2*(agg@W3) + b_post
// W_post rows: [0,64)=x  [64,320)=agg  [320,576)=agg*s1  [576,832)=agg*s2
// ---------------------------------------------------------------------------
__global__ void __launch_bounds__(128) post_gemm(
    const float* __restrict__ Xin, int ldx, const float* __restrict__ Agg,
    const float* __restrict__ S1, const float* __restrict__ S2,
    const float* __restrict__ Wp, const float* __restrict__ bp,
    float* __restrict__ Out) {
  __shared__ float Xs[16 * 68];
  __shared__ float Gs[16 * 260];
  __shared__ float S1s[16], S2s[16];
  const int tid = threadIdx.x;
  const int wave = tid >> 5, lane = tid & 31;
  const int m0 = blockIdx.x * 16;

  for (int i = tid; i < 16 * 64; i += 128) {
    int r = i >> 6, c = i & 63;
    Xs[r * 68 + c] = Xin[(m0 + r) * ldx + c];
  }
  for (int i = tid; i < 16 * 256; i += 128) {
    int r = i >> 8, c = i & 255;
    Gs[r * 260 + c] = Agg[(m0 + r) * 256 + c];
  }
  if (tid < 16) {
    S1s[tid] = S1[m0 + tid];
    S2s[tid] = S2[m0 + tid];
  }
  __syncthreads();

  const int n = lane & 15;
  const int kh = lane >> 4;
  const int col = wave * 16 + n;
  v8f acc0 = {}, acc2 = {}, acc3 = {};
#pragma unroll 4
  for (int k0 = 0; k0 < 64; k0 += 4) {   // x part -> acc0
    const int ka = k0 + 2 * kh;
    v2f a, b;
    a.x = Xs[n * 68 + ka];
    a.y = Xs[n * 68 + ka + 1];
    b.x = Wp[ka * 64 + col];
    b.y = Wp[(ka + 1) * 64 + col];
    acc0 = wmma_f32_k4(a, b, acc0);
  }
#pragma unroll 2
  for (int k0 = 0; k0 < 256; k0 += 4) {  // agg part -> acc0/acc2/acc3
    const int ka = k0 + 2 * kh;
    v2f a, b1, b2, b3;
    a.x = Gs[n * 260 + ka];
    a.y = Gs[n * 260 + ka + 1];
    const int r1 = (64 + ka) * 64 + col;
    const int r2 = (320 + ka) * 64 + col;
    const int r3 = (576 + ka) * 64 + col;
    b1.x = Wp[r1]; b1.y = Wp[r1 + 64];
    b2.x = Wp[r2]; b2.y = Wp[r2 + 64];
    b3.x = Wp[r3]; b3.y = Wp[r3 + 64];
    acc0 = wmma_f32_k4(a, b1, acc0);
    acc2 = wmma_f32_k4(a, b2, acc2);
    acc3 = wmma_f32_k4(a, b3, acc3);
  }
  const float bc = bp[col];
#pragma unroll
  for (int r = 0; r < 8; ++r) {
    const int row = r + 8 * kh;
    const float v = acc0[r] + S1s[row] * acc2[r] + S2s[row] * acc3[r] + bc;
    Out[(m0 + row) * 64 + col] = v;
  }
}

// ---------------------------------------------------------------------------
// CSR construction (once per launch; edge_index is constant across layers)
// ---------------------------------------------------------------------------
__global__ void zero_cnt(int* __restrict__ cnt) {
  int id = blockIdx.x * blockDim.x + threadIdx.x;
  if (id < N_NODES) cnt[id] = 0;
}

__global__ void count_edges(const int* __restrict__ ei, int* __restrict__ cnt) {
  int e = blockIdx.x * blockDim.x + threadIdx.x;
  if (e >= N_EDGES) return;
  atomicAdd(&cnt[ei[N_EDGES + e]], 1);
}

// Single-block exclusive scan of 50000 counts -> offs[50001]; fill = offs copy.
__global__ void __launch_bounds__(1024) scan_offsets(
    const int* __restrict__ cnt, int* __restrict__ offs,
    int* __restrict__ fill) {
  __shared__ int psum[1024];
  const int tid = threadIdx.x;
  const int CH = (N_NODES + 1023) / 1024;  // 49
  const int base = tid * CH;
  int s = 0;
  for (int i = 0; i < CH; ++i) {
    int idx = base + i;
    if (idx < N_NODES) s += cnt[idx];
  }
  psum[tid] = s;
  __syncthreads();
  for (int ofs = 1; ofs < 1024; ofs <<= 1) {  // Hillis-Steele inclusive scan
    int v = (tid >= ofs) ? psum[tid - ofs] : 0;
    __syncthreads();
    psum[tid] += v;
    __syncthreads();
  }
  int run = psum[tid] - s;  // exclusive prefix of this thread's chunk
  for (int i = 0; i < CH; ++i) {
    int idx = base + i;
    if (idx < N_NODES) {
      offs[idx] = run;
      fill[idx] = run;
      run += cnt[idx];
    }
  }
  if (tid == 1023) offs[N_NODES] = run;  // == N_EDGES
}

// Group src ids by dst: srcs[offs[dst] .. offs[dst+1]) = sources of dst's edges
__global__ void scatter_edges(const int* __restrict__ ei, int* __restrict__ fill,
                              int* __restrict__ srcs) {
  int e = blockIdx.x * blockDim.x + threadIdx.x;
  if (e >= N_EDGES) return;
  const int d = ei[N_EDGES + e];
  const int pos = atomicAdd(&fill[d], 1);
  srcs[pos] = ei[e];
}

// ---------------------------------------------------------------------------
// Atomic-free PNA aggregation, fused with finalize. 64 threads per node
// (one feature each; each wave is uniform over a single node). A[dst] term is
// node-constant; B[src] gathers are coalesced 256B rows that live in L2.
// ---------------------------------------------------------------------------
__global__ void __launch_bounds__(256) gather_reduce(
    const float* __restrict__ A, const float* __restrict__ B,
    const int* __restrict__ offs, const int* __restrict__ srcs,
    float* __restrict__ Agg, float* __restrict__ S1, float* __restrict__ S2) {
  const int tid = threadIdx.x;
  const int node = blockIdx.x * 4 + (tid >> 6);
  const int f = tid & 63;
  const int start = offs[node];
  const int end = offs[node + 1];
  const float a = A[node * 64 + f];  // b_pre already folded into A
  float s = 0.0f, q = 0.0f;
  float mn = __builtin_inff(), mx = -__builtin_inff();
  for (int e = start; e < end; ++e) {
    const int sidx = srcs[e];
    const float v = a + B[sidx * 64 + f];
    s += v;
    q += v * v;
    mn = fminf(mn, v);
    mx = fmaxf(mx, v);
  }
  const int degi = end - start;
  const float deg = fmaxf((float)degi, 1.0f);
  const float inv = 1.0f / deg;
  const float mean = s * inv;
  const float var = fmaxf(q * inv - mean * mean, 0.0f);
  const float sd = sqrtf(var + 1e-5f);
  const bool has = degi > 0;
  const int base = node * 256;
  Agg[base + f] = mean;
  Agg[base + 64 + f] = has ? mn : 0.0f;
  Agg[base + 128 + f] = has ? mx : 0.0f;
  Agg[base + 192 + f] = sd;
  if (f == 0) {
    const float logd = logf(deg + 1.0f);
    S1[node] = logd / AVG_DEG_LOG;
    S2[node] = AVG_DEG_LOG / logd;
  }
}

// ---------------------------------------------------------------------------
__global__ void copy_x(const float* __restrict__ x, float* __restrict__ out) {
  int id = blockIdx.x * blockDim.x + threadIdx.x;
  if (id >= N_NODES * 64) return;
  out[(id >> 6) * 256 + (id & 63)] = x[id];
}

__global__ void init_bn(float* __restrict__ bn) {
  if (threadIdx.x < 128) bn[threadIdx.x] = 0.0f;
}

// Per-feature sum / sumsq; stride is a multiple of 64 so each thread's feature
// index is fixed -> local accumulation, only 2 atomics per thread.
__global__ void bn_stats(const float* __restrict__ H, float* __restrict__ bn) {
  int id = blockIdx.x * blockDim.x + threadIdx.x;  // 65536 threads
  float s = 0.0f, q = 0.0f;
  for (int i = id; i < N_NODES * 64; i += 65536) {
    const float v = H[i];
    s += v;
    q += v * v;
  }
  const int f = id & 63;
  unsafeAtomicAdd(&bn[f], s);
  unsafeAtomicAdd(&bn[64 + f], q);
}

__global__ void bn_apply(const float* __restrict__ H,
                         const float* __restrict__ bn,
                         const float* __restrict__ gamma,
                         const float* __restrict__ beta,
                         float* __restrict__ out, int ldo) {
  int id = blockIdx.x * blockDim.x + threadIdx.x;
  if (id >= N_NODES * 64) return;
  const int nrow = id >> 6, f = id & 63;
  const float invN = 1.0f / (float)N_NODES;
  const float mu = bn[f] * invN;
  const float var = bn[64 + f] * invN - mu * mu;
  const float y = (H[id] - mu) * rsqrtf(var + BN_EPS) * gamma[f] + beta[f];
  out[nrow * ldo + f] = fmaxf(y, 0.0f);
}

// ---------------------------------------------------------------------------
extern "C" void kernel_launch(void* const* d_in, const int* in_sizes, int n_in,
                              void* d_out, int out_size, void* d_ws,
                              size_t ws_size, hipStream_t stream) {
  (void)in_sizes; (void)n_in; (void)out_size; (void)ws_size;
  const float* x = (const float*)d_in[0];
  const int* ei = (const int*)d_in[1];
  float* out = (float*)d_out;  // 50000 x 256, also holds per-layer inputs

  float* ws = (float*)d_ws;
  float* Abuf = ws;  ws += N_NODES * 64;
  float* Bbuf = ws;  ws += N_NODES * 64;
  float* Agg = ws;   ws += N_NODES * 256;
  float* S1 = ws;    ws += N_NODES;
  float* S2 = ws;    ws += N_NODES;
  float* tmp = ws;   ws += N_NODES * 64;
  float* hbuf = ws;  ws += N_NODES * 64;
  float* bn = ws;    ws += 128;
  int* cnt = (int*)ws;           // integer CSR scratch
  int* offs = cnt + N_NODES;     // N_NODES + 1
  int* fill = offs + N_NODES + 1;
  int* srcs = fill + N_NODES;    // N_EDGES

  const int EL = (N_NODES * 64 + 255) / 256;  // 12500
  const int GB = N_NODES / 16;                // 3125 (exact)
  const int NB = (N_NODES + 255) / 256;       // 196
  const int EB = (N_EDGES + 255) / 256;       // 3125
  const int AGB = N_NODES / 4;                // 12500 (4 nodes per block)

  // Build CSR once: edge_index is identical for all layers.
  zero_cnt<<<NB, 256, 0, stream>>>(cnt);
  count_edges<<<EB, 256, 0, stream>>>(ei, cnt);
  scan_offsets<<<1, 1024, 0, stream>>>(cnt, offs, fill);
  scatter_edges<<<EB, 256, 0, stream>>>(ei, fill, srcs);

  copy_x<<<EL, 256, 0, stream>>>(x, out);
  for (int l = 0; l < 3; ++l) {
    const float* Wpre  = (const float*)d_in[2 + 8 * l + 0];  // 128 x 64
    const float* bpre  = (const float*)d_in[2 + 8 * l + 1];
    const float* Wpost = (const float*)d_in[2 + 8 * l + 2];  // 832 x 64
    const float* bpost = (const float*)d_in[2 + 8 * l + 3];
    const float* Wlin  = (const float*)d_in[2 + 8 * l + 4];  // 64 x 64
    const float* blin  = (const float*)d_in[2 + 8 * l + 5];
    const float* gamma = (const float*)d_in[2 + 8 * l + 6];
    const float* beta  = (const float*)d_in[2 + 8 * l + 7];
    const float* in_f = out + 64 * l;  // layer input slice, row stride 256

    // A = x@W_pre[0:64] + b_pre ; B = x@W_pre[64:128]
    gemm64<<<GB, 128, 0, stream>>>(in_f, 256, Wpre, bpre, Abuf, 64);
    gemm64<<<GB, 128, 0, stream>>>(in_f, 256, Wpre + 64 * 64, nullptr, Bbuf, 64);
    gather_reduce<<<AGB, 256, 0, stream>>>(Abuf, Bbuf, offs, srcs, Agg, S1, S2);
    post_gemm<<<GB, 128, 0, stream>>>(in_f, 256, Agg, S1, S2, Wpost, bpost, tmp);
    gemm64<<<GB, 128, 0, stream>>>(tmp, 64, Wlin, blin, hbuf, 64);
    init_bn<<<1, 128, 0, stream>>>(bn);
    bn_stats<<<256, 256, 0, stream>>>(hbuf, bn);
    bn_apply<<<EL, 256, 0, stream>>>(hbuf, bn, gamma, beta, out + 64 * (l + 1), 256);
  }
}